// MS3DA_16028817949562
// MI455X (gfx1250) — compile-verified
//
#include <hip/hip_runtime.h>
#include <cstdint>

typedef float v2f __attribute__((ext_vector_type(2)));
typedef float v4f __attribute__((ext_vector_type(4)));
typedef float v8f __attribute__((ext_vector_type(8)));

#define BB 32
#define CC 256
#define HH 64
#define WW 64
#define LSTR 68   // padded LDS row stride (floats): 272 B rows, 16 B aligned, spreads banks

// ---------------------------------------------------------------------------
// Kernel 1: per-(b,c) plane reductions.
//   - stage 64x64 fp32 plane (16 KB) into LDS via CDNA5 async global->LDS DMA
//   - row means (x_h) and column means (x_w) via V_WMMA_F32_16X16X4_F32 with
//     an all-ones B matrix (sum-by-matmul, full fp32 precision)
//   - c_in = plane mean
// ---------------------------------------------------------------------------
__global__ __launch_bounds__(128) void ms3da_reduce(const float* __restrict__ x,
                                                    float* __restrict__ xh,
                                                    float* __restrict__ xw,
                                                    float* __restrict__ cin)
{
    __shared__ float tile[HH * LSTR];
    __shared__ float rowm[HH];
    __shared__ float colm[WW];

    const int plane = blockIdx.x;                     // b*C + c
    const float* __restrict__ src = x + (size_t)plane * (HH * WW);
    const int t = threadIdx.x;                        // 0..127 (4 waves)

    // ---- async stage: 1024 x float4, ASYNCcnt-tracked -------------------
    for (int j = 0; j < 8; ++j) {
        const int i4 = t + 128 * j;                   // float4 index 0..1023
        const int r  = i4 >> 4;
        const int c4 = i4 & 15;
        const unsigned loff =
            (unsigned)(uintptr_t)(&tile[r * LSTR + c4 * 4]);
        const unsigned long long gaddr =
            (unsigned long long)(uintptr_t)(src + r * WW + c4 * 4);
        asm volatile("global_load_async_to_lds_b128 %0, %1, off"
                     :: "v"(loff), "v"(gaddr) : "memory");
    }
    asm volatile("s_wait_asynccnt 0" ::: "memory");
    __syncthreads();

    const int lane = t & 31;
    const int wv   = t >> 5;       // wave 0..3: 16-row / 16-col strip each
    const int m    = lane & 15;    // A-matrix row within strip
    const int hi   = lane >> 4;    // lanes 16..31 carry K=2,3 of the 16x4 tile

    v2f ones; ones[0] = 1.0f; ones[1] = 1.0f;

    // ---- row sums: A = plane[16-row strip][4-col slice], B = ones -------
    {
        v8f acc = {};
        const int row = 16 * wv + m;
        for (int kb = 0; kb < 16; ++kb) {
            v2f a = *(const v2f*)&tile[row * LSTR + kb * 4 + hi * 2];
            acc = __builtin_amdgcn_wmma_f32_16x16x4_f32(
                      false, a, false, ones, (short)0, acc, false, false);
        }
        // D layout: VGPR r, lanes 0-15 -> M=r ; lanes 16-31 -> M=8+r
        if (m == 0) {
            #pragma unroll
            for (int r = 0; r < 8; ++r)
                rowm[16 * wv + 8 * hi + r] = acc[r] * (1.0f / 64.0f);
        }
    }

    // ---- col sums: transposed A reads from padded LDS -------------------
    {
        v8f acc = {};
        const int col = 16 * wv + m;
        for (int kb = 0; kb < 16; ++kb) {
            const int k0 = kb * 4 + hi * 2;
            v2f a;
            a[0] = tile[(k0 + 0) * LSTR + col];
            a[1] = tile[(k0 + 1) * LSTR + col];
            acc = __builtin_amdgcn_wmma_f32_16x16x4_f32(
                      false, a, false, ones, (short)0, acc, false, false);
        }
        if (m == 0) {
            #pragma unroll
            for (int r = 0; r < 8; ++r)
                colm[16 * wv + 8 * hi + r] = acc[r] * (1.0f / 64.0f);
        }
    }
    __syncthreads();

    if (t < HH)               xh[(size_t)plane * HH + t]        = rowm[t];
    else if (t < HH + WW)     xw[(size_t)plane * WW + (t - HH)] = colm[t - HH];
    if (t == 0) {
        float s = 0.0f;
        for (int h = 0; h < HH; ++h) s += rowm[h];
        cin[plane] = s * (1.0f / 64.0f);              // mean over H of x_h
    }
}

// ---------------------------------------------------------------------------
// Kernel 2: per-batch gates (tiny).
//   emb[j] = mean_c [x_h;x_w][b,c,j] (j<128); emb = corr(emb,wsp1)+corr(emb,wsp2)
//   h_emb = emb[:64], w_emb = emb[64:]
//   cch[c] = corr(c_in,wc1)+corr(c_in,wc2)+corr(c_in,wc3)  (length C=256)
// (XLA conv_general_dilated is cross-correlation: out[j] = sum_k w[k]*in[j+k-K/2])
// ---------------------------------------------------------------------------
__global__ __launch_bounds__(256) void ms3da_gate(const float* __restrict__ xh,
                                                  const float* __restrict__ xw,
                                                  const float* __restrict__ cin,
                                                  const float* __restrict__ wsp1,
                                                  const float* __restrict__ wsp2,
                                                  const float* __restrict__ wc1,
                                                  const float* __restrict__ wc2,
                                                  const float* __restrict__ wc3,
                                                  float* __restrict__ hemb,
                                                  float* __restrict__ wemb,
                                                  float* __restrict__ cch)
{
    const int b = blockIdx.x;
    const int t = threadIdx.x;
    __shared__ float emb[2 * HH];
    __shared__ float cv[CC];

    if (t < 2 * HH) {
        const float* base = (t < HH) ? (xh + (size_t)(b * CC) * HH + t)
                                     : (xw + (size_t)(b * CC) * WW + (t - HH));
        float s = 0.0f;
        for (int c = 0; c < CC; ++c) s += base[(size_t)c * HH];
        emb[t] = s * (1.0f / CC);
    }
    cv[t] = cin[(size_t)b * CC + t];
    __syncthreads();

    if (t < 2 * HH) {
        float s = 0.0f;
        #pragma unroll
        for (int k = 0; k < 3; ++k) {
            const int j = t + k - 1;
            if (0 <= j && j < 2 * HH) s += wsp1[k] * emb[j];
        }
        #pragma unroll
        for (int k = 0; k < 5; ++k) {
            const int j = t + k - 2;
            if (0 <= j && j < 2 * HH) s += wsp2[k] * emb[j];
        }
        if (t < HH) hemb[b * HH + t]        = s;
        else        wemb[b * WW + (t - HH)] = s;
    }
    {
        float s = 0.0f;
        #pragma unroll
        for (int k = 0; k < 3; ++k) {
            const int j = t + k - 1;
            if (0 <= j && j < CC) s += wc1[k] * cv[j];
        }
        #pragma unroll
        for (int k = 0; k < 5; ++k) {
            const int j = t + k - 2;
            if (0 <= j && j < CC) s += wc2[k] * cv[j];
        }
        #pragma unroll
        for (int k = 0; k < 5; ++k) {
            const int j = t + k - 2;
            if (0 <= j && j < CC) s += wc3[k] * cv[j];
        }
        cch[(size_t)b * CC + t] = s;
    }
}

// ---------------------------------------------------------------------------
// Kernel 3: fused BN + sigmoid gate, float4-vectorized, NT stores so the
// streamed output does not evict x (which fits in the 192 MB L2) mid-pass.
// ---------------------------------------------------------------------------
__global__ __launch_bounds__(256) void ms3da_apply(const float* __restrict__ x,
                                                   const float* __restrict__ hemb,
                                                   const float* __restrict__ wemb,
                                                   const float* __restrict__ cch,
                                                   const float* __restrict__ gamma,
                                                   const float* __restrict__ beta,
                                                   const float* __restrict__ rmean,
                                                   const float* __restrict__ rvar,
                                                   float* __restrict__ out)
{
    const unsigned i4 = blockIdx.x * blockDim.x + threadIdx.x; // float4 index
    const int w4 = i4 & 15;
    const int h  = (i4 >> 4)  & 63;
    const int c  = (i4 >> 10) & 255;
    const int b  = (int)(i4 >> 18);

    const v4f xv = *(const v4f*)(x    + (size_t)i4 * 4);
    const v4f wv = *(const v4f*)(wemb + b * WW + w4 * 4);
    const float g  = gamma[c] * rsqrtf(rvar[c] + 1e-5f);
    const float a  = hemb[b * HH + h] * cch[b * CC + c] * g;
    const float mb = beta[c] - rmean[c] * g;

    v4f o;
    #pragma unroll
    for (int e = 0; e < 4; ++e) {
        const float z = a * wv[e] + mb;            // bn value
        o[e] = xv[e] / (1.0f + __expf(-z));        // x * sigmoid(bn)
    }
    __builtin_nontemporal_store(o, (v4f*)(out + (size_t)i4 * 4));
}

// ---------------------------------------------------------------------------
extern "C" void kernel_launch(void* const* d_in, const int* in_sizes, int n_in,
                              void* d_out, int out_size, void* d_ws, size_t ws_size,
                              hipStream_t stream)
{
    const float* x     = (const float*)d_in[0];
    const float* wsp1  = (const float*)d_in[1];
    const float* wsp2  = (const float*)d_in[2];
    const float* wc1   = (const float*)d_in[3];
    const float* wc2   = (const float*)d_in[4];
    const float* wc3   = (const float*)d_in[5];
    const float* gamma = (const float*)d_in[6];
    const float* beta  = (const float*)d_in[7];
    const float* rmean = (const float*)d_in[8];
    const float* rvar  = (const float*)d_in[9];

    float* ws   = (float*)d_ws;                 // all offsets 16 B aligned
    float* xh   = ws;                           // B*C*H   = 524288
    float* xw   = ws + 524288;                  // B*C*W   = 524288
    float* cin  = ws + 1048576;                 // B*C     = 8192
    float* hemb = ws + 1056768;                 // B*H     = 2048
    float* wemb = ws + 1058816;                 // B*W     = 2048
    float* cch  = ws + 1060864;                 // B*C     = 8192

    ms3da_reduce<<<BB * CC, 128, 0, stream>>>(x, xh, xw, cin);
    ms3da_gate<<<BB, CC, 0, stream>>>(xh, xw, cin, wsp1, wsp2, wc1, wc2, wc3,
                                      hemb, wemb, cch);
    const unsigned n4 = (unsigned)(BB * CC * HH * WW) / 4;   // 8388608
    ms3da_apply<<<n4 / 256, 256, 0, stream>>>(x, hemb, wemb, cch,
                                              gamma, beta, rmean, rvar,
                                              (float*)d_out);
}